// LLMEmbedding_82094004896325
// MI455X (gfx1250) — compile-verified
//
#include <hip/hip_runtime.h>
#include <math.h>

typedef float v2f __attribute__((ext_vector_type(2)));
typedef float v4f __attribute__((ext_vector_type(4)));
typedef float v8f __attribute__((ext_vector_type(8)));

#define N_NEI    10
#define MEM_DIM  172
#define TOK_DIM  4096
#define TIME_DIM 100
#define BATCH    16
#define LEN_TOK  512

#define ROWS_MEM   (BATCH * (1 + N_NEI))      // 176 rows per gemm (mem1 / mem2)
#define ROWS_TIME  (BATCH * (1 + 2 * N_NEI))  // 336 rows of time encoding
#define CAT_ROWS   (BATCH * 2 * (1 + N_NEI))  // 352 rows in cat_mem

// ---------------------------------------------------------------------------
// Token -> (IDX_SRC, IDX_DST, IDX_TIME), closed form of the reference's
// _build_indices() with P=(14,21,14,21,20), N_NEI=10, off2=11.
// Segments: [0,15) [15,225) [225,239) [239,449) [449,469) [469,512)
// ---------------------------------------------------------------------------
__device__ __forceinline__ void token_indices(int t, int& s, int& d, int& ti) {
  if (t < 15)       { s = 0;  d = (t == 0) ? 11 : 0; ti = 0; }
  else if (t < 225) { int j = (t - 15)  / 21; s = 0;  d = 12 + j; ti = 1 + j;  }
  else if (t < 239) { s = 11; d = 11; ti = 0; }
  else if (t < 449) { int j = (t - 239) / 21; s = 11; d = 1 + j;  ti = 11 + j; }
  else if (t < 469) { s = 11; d = 11; ti = 0; }
  else              { s = 0;  d = 11; ti = 0; }
}

// ---------------------------------------------------------------------------
// tenc[g][k] = cos((ts[b] - t_ago[b,r]) * w_time[k] + phase_time[k])
// g = b*21 + r ; r==0 -> timestamp itself, r in [1,10] -> src_edge_times,
// r in [11,20] -> dst_edge_times
// ---------------------------------------------------------------------------
__global__ __launch_bounds__(256) void time_enc_kernel(
    const float* __restrict__ timestamps,
    const float* __restrict__ src_et,
    const float* __restrict__ dst_et,
    const float* __restrict__ w_time,
    const float* __restrict__ phase_time,
    float*       __restrict__ tenc)
{
  int idx = blockIdx.x * 256 + threadIdx.x;
  if (idx >= ROWS_TIME * TIME_DIM) return;
  int k = idx % TIME_DIM;
  int g = idx / TIME_DIM;
  int b = g / (1 + 2 * N_NEI);
  int r = g - b * (1 + 2 * N_NEI);
  float ts = timestamps[b];
  float ta = (r == 0) ? ts
           : (r <= N_NEI ? src_et[b * N_NEI + (r - 1)]
                         : dst_et[b * N_NEI + (r - 1 - N_NEI)]);
  float delta = ts - ta;
  tenc[idx] = cosf(fmaf(delta, w_time[k], phase_time[k]));
}

// ---------------------------------------------------------------------------
// cat_mem[b*22 + which*11 + r][:] = memory[node] @ W + bias
// WMMA f32 16x16x4. A tile (16 gathered memory rows x 172) staged in LDS.
// 8 waves/block; each wave produces two adjacent 16x16 output tiles.
// A fragment per lane: m = lane&15, kpair = (lane>>4)*2, a = A[m][k0+kp .. +1]
// B fragment per lane: n = lane&15, same kpair selection on rows of W.
// C: vgpr i holds M = i + (lane>>4)*8, N = lane&15.
// ---------------------------------------------------------------------------
__global__ __launch_bounds__(256) void gemm_mem_kernel(
    const float* __restrict__ memory,
    const int*   __restrict__ src_nodes,
    const int*   __restrict__ dst_nodes,
    const int*   __restrict__ src_nei,
    const int*   __restrict__ dst_nei,
    const float* __restrict__ W1, const float* __restrict__ b1,
    const float* __restrict__ W2, const float* __restrict__ b2,
    float*       __restrict__ cat_mem)
{
  const int which = blockIdx.z;                    // 0: mem1 path, 1: mem2 path
  const float* __restrict__ W    = which ? W2 : W1;
  const float* __restrict__ bias = which ? b2 : b1;
  const int*   __restrict__ head = which ? dst_nodes : src_nodes;
  const int*   __restrict__ nei  = which ? src_nei   : dst_nei;
  const int mtile = blockIdx.y;                    // 0..10

  __shared__ float As[16][MEM_DIM];
  for (int i = threadIdx.x; i < 16 * MEM_DIM; i += 256) {
    int m = i / MEM_DIM, k = i - m * MEM_DIM;
    int g = mtile * 16 + m;                        // 0..175
    int b = g / (1 + N_NEI), r = g - b * (1 + N_NEI);
    int node = (r == 0) ? head[b] : nei[b * N_NEI + (r - 1)];
    As[m][k] = memory[(long)node * MEM_DIM + k];
  }
  __syncthreads();

  const int wave  = threadIdx.x >> 5;
  const int lane  = threadIdx.x & 31;
  const int half  = lane >> 4;
  const int m     = lane & 15;
  const int kh    = half * 2;
  const int ncol0 = blockIdx.x * 256 + wave * 32 + (lane & 15);
  const int ncol1 = ncol0 + 16;

  v8f acc0 = {}, acc1 = {};
  for (int k0 = 0; k0 < MEM_DIM; k0 += 4) {        // 43 exact steps
    v2f a, bv0, bv1;
    a.x = As[m][k0 + kh];
    a.y = As[m][k0 + kh + 1];
    const float* Wr0 = W + (long)(k0 + kh)     * TOK_DIM;
    const float* Wr1 = W + (long)(k0 + kh + 1) * TOK_DIM;
    bv0.x = Wr0[ncol0]; bv0.y = Wr1[ncol0];
    bv1.x = Wr0[ncol1]; bv1.y = Wr1[ncol1];
    acc0 = __builtin_amdgcn_wmma_f32_16x16x4_f32(false, a, false, bv0,
                                                 (short)0, acc0, false, false);
    acc1 = __builtin_amdgcn_wmma_f32_16x16x4_f32(false, a, false, bv1,
                                                 (short)0, acc1, false, false);
  }

  const float bb0 = bias[ncol0];
  const float bb1 = bias[ncol1];
#pragma unroll
  for (int i = 0; i < 8; i++) {
    int g = mtile * 16 + i + half * 8;
    int b = g / (1 + N_NEI), r = g - b * (1 + N_NEI);
    long row = (long)(b * 22 + which * (1 + N_NEI) + r) * TOK_DIM;
    cat_mem[row + ncol0] = acc0[i] + bb0;
    cat_mem[row + ncol1] = acc1[i] + bb1;
  }
}

// ---------------------------------------------------------------------------
// texp[g][:] = tenc[g] @ Wt + bt   (336 x 100 x 4096), same WMMA scheme.
// ---------------------------------------------------------------------------
__global__ __launch_bounds__(256) void gemm_time_kernel(
    const float* __restrict__ tenc,
    const float* __restrict__ Wt, const float* __restrict__ bt,
    float*       __restrict__ texp)
{
  const int mtile = blockIdx.y;                    // 0..20

  __shared__ float As[16][TIME_DIM];
  for (int i = threadIdx.x; i < 16 * TIME_DIM; i += 256) {
    int m = i / TIME_DIM, k = i - m * TIME_DIM;
    As[m][k] = tenc[(long)(mtile * 16 + m) * TIME_DIM + k];
  }
  __syncthreads();

  const int wave  = threadIdx.x >> 5;
  const int lane  = threadIdx.x & 31;
  const int half  = lane >> 4;
  const int m     = lane & 15;
  const int kh    = half * 2;
  const int ncol0 = blockIdx.x * 256 + wave * 32 + (lane & 15);
  const int ncol1 = ncol0 + 16;

  v8f acc0 = {}, acc1 = {};
  for (int k0 = 0; k0 < TIME_DIM; k0 += 4) {       // 25 exact steps
    v2f a, bv0, bv1;
    a.x = As[m][k0 + kh];
    a.y = As[m][k0 + kh + 1];
    const float* Wr0 = Wt + (long)(k0 + kh)     * TOK_DIM;
    const float* Wr1 = Wt + (long)(k0 + kh + 1) * TOK_DIM;
    bv0.x = Wr0[ncol0]; bv0.y = Wr1[ncol0];
    bv1.x = Wr0[ncol1]; bv1.y = Wr1[ncol1];
    acc0 = __builtin_amdgcn_wmma_f32_16x16x4_f32(false, a, false, bv0,
                                                 (short)0, acc0, false, false);
    acc1 = __builtin_amdgcn_wmma_f32_16x16x4_f32(false, a, false, bv1,
                                                 (short)0, acc1, false, false);
  }

  const float bb0 = bt[ncol0];
  const float bb1 = bt[ncol1];
#pragma unroll
  for (int i = 0; i < 8; i++) {
    long row = (long)(mtile * 16 + i + half * 8) * TOK_DIM;
    texp[row + ncol0] = acc0[i] + bb0;
    texp[row + ncol1] = acc1[i] + bb1;
  }
}

// ---------------------------------------------------------------------------
// Streaming expansion: 402 MB of nontemporal b128 stores, sources stay in L2.
// idx -> (b, t, d4): d4 fastest so a wave writes 512B contiguous per tensor.
// ---------------------------------------------------------------------------
__global__ __launch_bounds__(256) void expand_kernel(
    const v4f* __restrict__ cat_mem,
    const v4f* __restrict__ texp,
    v4f*       __restrict__ out)
{
  const long R4     = TOK_DIM / 4;                   // 1024 float4 per row
  const long STRIDE = (long)BATCH * LEN_TOK * R4;    // one output tensor
  long idx = (long)blockIdx.x * 256 + threadIdx.x;
  int d4 = (int)(idx & (R4 - 1));
  int t  = (int)((idx >> 10) & (LEN_TOK - 1));
  int b  = (int)(idx >> 19);
  int s, d, ti;
  token_indices(t, s, d, ti);
  v4f v0 = cat_mem[(long)(b * 22 + s)  * R4 + d4];
  v4f v1 = cat_mem[(long)(b * 22 + d)  * R4 + d4];
  v4f v2 = texp   [(long)(b * 21 + ti) * R4 + d4];
  long o = ((long)b * LEN_TOK + t) * R4 + d4;
  __builtin_nontemporal_store(v0, out + o);
  __builtin_nontemporal_store(v1, out + o + STRIDE);
  __builtin_nontemporal_store(v2, out + o + 2 * STRIDE);
}

// ---------------------------------------------------------------------------
extern "C" void kernel_launch(void* const* d_in, const int* in_sizes, int n_in,
                              void* d_out, int out_size, void* d_ws, size_t ws_size,
                              hipStream_t stream) {
  const float* memory     = (const float*)d_in[0];
  const int*   src_nodes  = (const int*)  d_in[1];
  const int*   dst_nodes  = (const int*)  d_in[2];
  const int*   src_nei    = (const int*)  d_in[3];
  const int*   dst_nei    = (const int*)  d_in[4];
  const float* timestamps = (const float*)d_in[5];
  const float* src_et     = (const float*)d_in[6];
  const float* dst_et     = (const float*)d_in[7];
  const float* W1         = (const float*)d_in[8];
  const float* b1         = (const float*)d_in[9];
  const float* W2         = (const float*)d_in[10];
  const float* b2         = (const float*)d_in[11];
  const float* Wt         = (const float*)d_in[12];
  const float* bt         = (const float*)d_in[13];
  const float* w_time     = (const float*)d_in[14];
  const float* phase      = (const float*)d_in[15];

  // Workspace layout (all offsets 16B aligned):
  float* tenc    = (float*)d_ws;                          // 336*100
  float* cat_mem = tenc + (long)ROWS_TIME * TIME_DIM;     // 352*4096
  float* texp    = cat_mem + (long)CAT_ROWS * TOK_DIM;    // 336*4096

  time_enc_kernel<<<(ROWS_TIME * TIME_DIM + 255) / 256, 256, 0, stream>>>(
      timestamps, src_et, dst_et, w_time, phase, tenc);

  dim3 g1(TOK_DIM / 256, ROWS_MEM / 16, 2);
  gemm_mem_kernel<<<g1, 256, 0, stream>>>(
      memory, src_nodes, dst_nodes, src_nei, dst_nei, W1, b1, W2, b2, cat_mem);

  dim3 g2(TOK_DIM / 256, ROWS_TIME / 16, 1);
  gemm_time_kernel<<<g2, 256, 0, stream>>>(tenc, Wt, bt, texp);

  long total4 = (long)BATCH * LEN_TOK * (TOK_DIM / 4);    // 8,388,608 threads
  expand_kernel<<<(unsigned)(total4 / 256), 256, 0, stream>>>(
      (const v4f*)cat_mem, (const v4f*)texp, (v4f*)d_out);
}